// SDTBoundaryMask_74981539053937
// MI455X (gfx1250) — compile-verified
//
#include <hip/hip_runtime.h>
#include <hip/hip_bf16.h>
#include <stdint.h>

// ---------------------------------------------------------------------------
// 3-D exact EDT boundary mask for a 192^3 binary volume (MI455X / gfx1250).
// Memory-bound: ~350MB of traffic -> ~15us at 23.3 TB/s; 2x28MB scratch lives
// in the 192MB L2. Min-plus parabola transform cannot map to WMMA; the CDNA5
// features used are the Tensor Data Mover (tensor_load_to_lds /
// tensor_store_from_lds) for strided 2-D tile movement on the transposed axis
// passes, TENSORcnt waits, and LDS-resident wave32 compute.
// ---------------------------------------------------------------------------

#define N 192
#define N2 (N * N)
#define N3 (N * N * N)
#define SENT 10000.0f // sentinel city-block distance; SENT^2 == 1e8 == LARGE

typedef uint32_t u32;
typedef u32 v4u __attribute__((ext_vector_type(4)));
typedef int v4i __attribute__((ext_vector_type(4)));
typedef int v8i __attribute__((ext_vector_type(8)));

#if defined(__HIP_DEVICE_COMPILE__) && __has_builtin(__builtin_amdgcn_tensor_load_to_lds)
#define HAVE_TDM 1
#else
#define HAVE_TDM 0
#endif

#if HAVE_TDM
struct TdmDesc {
  v4u g0;
  v8i g1;
};

// 2-D tile descriptor: tile_dim0 = 16 contiguous f32, tile_dim1 = 192 rows,
// row stride S elements. Built from wave-uniform values, pinned to SGPRs.
__device__ __forceinline__ TdmDesc tdm_make_desc(const float* gptr, u32 ldsAddr,
                                                 int S) {
  uint64_t ga = (uint64_t)(uintptr_t)gptr;
  // D# group 0: count=1 | lds_addr | global_addr[56:0] | type=2
  u32 g00 = 1u;
  u32 g01 = ldsAddr;
  u32 g02 = (u32)ga;
  u32 g03 = (u32)((ga >> 32) & 0x01FFFFFFu) | (2u << 30);
  // D# group 1: data_size=4B; tensor_dim0=16, tensor_dim1=192;
  // tile_dim0=16, tile_dim1=192; tensor_dim0_stride=S (elements)
  u32 g10 = (2u << 16);     // workgroup_mask=0, data_size=2 (4 bytes)
  u32 g11 = (16u << 16);    // tensor_dim0[15:0]
  u32 g12 = ((u32)N << 16); // tensor_dim0 hi=0 | tensor_dim1[15:0]
  u32 g13 = (16u << 16);    // tensor_dim1 hi=0 | tile_dim0
  u32 g14 = (u32)N;         // tile_dim1 | tile_dim2=0
  u32 g15 = (u32)S;         // tensor_dim0_stride[31:0]
#define RFL(v) __builtin_amdgcn_readfirstlane((int)(v))
  TdmDesc d;
  d.g0 = (v4u){(u32)RFL(g00), (u32)RFL(g01), (u32)RFL(g02), (u32)RFL(g03)};
  d.g1 = (v8i){RFL(g10), RFL(g11), RFL(g12), RFL(g13),
               RFL(g14), RFL(g15), 0, 0};
#undef RFL
  return d;
}
#endif

// ---------------------------------------------------------------------------
// Generic 1-D parabola min-plus pass along a strided axis, in place.
// Tile = 192 rows (DT axis, element stride S) x 16 contiguous columns.
// Wave 0 pulls the tile into LDS via TDM, waits on TENSORcnt; all 8 waves
// compute out[i] = min_j f[j] + (i-j)^2 from LDS; wave 0 pushes the result
// tile back with tensor_store_from_lds (sequential DMA instead of a
// stride-S 4-byte scatter).
// ---------------------------------------------------------------------------
__global__ __launch_bounds__(256) void k_dt_pass(float* __restrict__ vol,
                                                 int S, int P) {
  __shared__ float fin[N * 16];  // [j][c], row length 16 -> conflict-free
  __shared__ float fout[N * 16]; // [i][c]

  const int c0 = blockIdx.x * 16; // contiguous-axis chunk
  const int plane = blockIdx.y;   // orthogonal plane index
  const int baseElem = plane * P + c0;

#if HAVE_TDM
  if (threadIdx.x < 32) { // wave 0 only: TDM is a wave-level DMA (EXEC ignored)
    TdmDesc d = tdm_make_desc(vol + baseElem, (u32)(uintptr_t)&fin[0], S);
    v4i z4 = {0, 0, 0, 0};
    v8i z8 = {0, 0, 0, 0, 0, 0, 0, 0};
    __builtin_amdgcn_tensor_load_to_lds(d.g0, d.g1, z4, z4, z8, 0);
    __builtin_amdgcn_s_wait_tensorcnt(0);
  }
  __syncthreads();
#else
  for (int k = threadIdx.x; k < N * 16; k += 256) {
    int j = k >> 4, cc = k & 15;
    fin[k] = vol[baseElem + j * S + cc];
  }
  __syncthreads();
#endif

  const int c = threadIdx.x & 15;  // column within tile
  const int it = threadIdx.x >> 4; // 0..15; thread owns i = it + 16*r
  float acc[12];
#pragma unroll
  for (int r = 0; r < 12; ++r) acc[r] = 3.0e38f;

  for (int j = 0; j < N; j += 2) {
    float s0 = fin[j * 16 + c]; // LDS broadcast, conflict-free
    float s1 = fin[j * 16 + 16 + c];
    float t0 = (float)(it - j);
#pragma unroll
    for (int r = 0; r < 12; ++r) {
      float ta = t0 + (float)(16 * r);
      float tb = ta - 1.0f;
      float ca = __builtin_fmaf(ta, ta, s0);
      float cb = __builtin_fmaf(tb, tb, s1);
      acc[r] = fminf(acc[r], fminf(ca, cb)); // -> v_min3_num_f32
    }
  }

#pragma unroll
  for (int r = 0; r < 12; ++r) fout[(it + 16 * r) * 16 + c] = acc[r];
  __syncthreads();

#if HAVE_TDM
  if (threadIdx.x < 32) {
    TdmDesc d = tdm_make_desc(vol + baseElem, (u32)(uintptr_t)&fout[0], S);
    v4i z4 = {0, 0, 0, 0};
    v8i z8 = {0, 0, 0, 0, 0, 0, 0, 0};
    __builtin_amdgcn_tensor_store_from_lds(d.g0, d.g1, z4, z4, z8, 0);
    __builtin_amdgcn_s_wait_tensorcnt(0);
  }
#else
  float* outp = vol + baseElem + c;
#pragma unroll
  for (int r = 0; r < 12; ++r) {
    int i = it + 16 * r;
    outp[i * S] = acc[r];
  }
#endif
}

// ---------------------------------------------------------------------------
// Pass 1 (axis 2, contiguous lines), fused with init. Binary input -> exact
// 1-D DT via two O(n) city-block scans, then square. Produces A = dist^2 for
// fg mask and B = dist^2 for bg mask. One thread owns one full line.
// ---------------------------------------------------------------------------
__global__ __launch_bounds__(256) void k_init_dt_z(const float* __restrict__ x,
                                                   float* __restrict__ A,
                                                   float* __restrict__ B) {
  int line = blockIdx.x * blockDim.x + threadIdx.x;
  if (line >= N2) return;
  const float* p = x + (size_t)line * N;
  float* a = A + (size_t)line * N;
  float* b = B + (size_t)line * N;

  float da = SENT, db = SENT;
  for (int k = 0; k < N; ++k) {
    float v = p[k];
    bool fg = (v != 0.0f);
    da = fg ? fminf(da + 1.0f, SENT) : 0.0f;
    db = fg ? 0.0f : fminf(db + 1.0f, SENT);
    a[k] = da;
    b[k] = db;
  }
  da = SENT;
  db = SENT;
  for (int k = N - 1; k >= 0; --k) {
    da = fminf(da + 1.0f, a[k]); // both operands <= SENT+1; min caps at SENT
    db = fminf(db + 1.0f, b[k]);
    da = fminf(da, SENT);
    db = fminf(db, SENT);
    a[k] = da * da; // == 1e8 (LARGE) when no opposite voxel in this line
    b[k] = db * db;
  }
}

// ---------------------------------------------------------------------------
// Finalize: out = (sqrt(A) + sqrt(B)) < border_dist ? 1 : 0
// ---------------------------------------------------------------------------
__global__ __launch_bounds__(256) void k_final(const float* __restrict__ A,
                                               const float* __restrict__ B,
                                               float* __restrict__ out) {
  int idx = blockIdx.x * 256 + threadIdx.x;
  if (idx >= N3) return;
  int i2 = idx % N;
  int t = idx / N;
  int i1 = t % N;
  int i0 = t / N;
  float b0 = fminf((float)(i0 + 1), (float)(N - i0));
  float b1 = fminf((float)(i1 + 1), (float)(N - i1));
  float b2 = fminf((float)(i2 + 1), (float)(N - i2));
  float border = fminf(b0, fminf(b1, b2));
  float d = sqrtf(A[idx]) + sqrtf(B[idx]);
  out[idx] = (d < border) ? 1.0f : 0.0f;
}

// ---------------------------------------------------------------------------
extern "C" void kernel_launch(void* const* d_in, const int* in_sizes, int n_in,
                              void* d_out, int out_size, void* d_ws,
                              size_t ws_size, hipStream_t stream) {
  (void)in_sizes;
  (void)n_in;
  (void)out_size;
  (void)ws_size;
  const float* x = (const float*)d_in[0];
  float* out = (float*)d_out;
  float* A = (float*)d_ws;   // fg squared distances
  float* B = A + (size_t)N3; // bg squared distances (2*28.3MB workspace)

  // Pass 1: init + DT along contiguous axis (i2), both masks.
  k_init_dt_z<<<(N2 + 255) / 256, 256, 0, stream>>>(x, A, B);

  dim3 grid2(N / 16, N);
  // Pass 2: DT along axis 1 (stride 192), in place.
  k_dt_pass<<<grid2, 256, 0, stream>>>(A, N, N2);
  k_dt_pass<<<grid2, 256, 0, stream>>>(B, N, N2);
  // Pass 3: DT along axis 0 (stride 192^2), in place.
  k_dt_pass<<<grid2, 256, 0, stream>>>(A, N2, N);
  k_dt_pass<<<grid2, 256, 0, stream>>>(B, N2, N);

  // Finalize boundary mask.
  k_final<<<(N3 + 255) / 256, 256, 0, stream>>>(A, B, out);
}